// GATlayer_70282844831795
// MI455X (gfx1250) — compile-verified
//
#include <hip/hip_runtime.h>
#include <math.h>

#define HEADS     2
#define OUT_CH    128
#define IN_CH     128
#define N_NODES   50000
#define N_EDGES   800000
#define NEG_SLOPE 0.2f
#define TOT_CH    (HEADS * OUT_CH)   // 256

typedef float v2f __attribute__((ext_vector_type(2)));
typedef float v8f __attribute__((ext_vector_type(8)));

__device__ __forceinline__ float lrelu(float v) { return v > 0.0f ? v : NEG_SLOPE * v; }

// sign-aware float atomic max via integer atomics (works with -inf init)
__device__ __forceinline__ void atomicMaxF(float* a, float v) {
  if (v >= 0.0f) atomicMax((int*)a, __float_as_int(v));
  else           atomicMin((unsigned int*)a, __float_as_uint(v));
}

// ---------------------------------------------------------------------------
// Kernel 1: X[N,256] = A[N,128] @ B[128,256] via V_WMMA_F32_16X16X4_F32.
// One block = 16-row M-panel; 8 waves, each wave owns two 16x16 N-tiles.
// ---------------------------------------------------------------------------
__global__ __launch_bounds__(256) void gat_gemm(const float* __restrict__ A,
                                                const float* __restrict__ B,
                                                float* __restrict__ X) {
  __shared__ float lds[16 * 132];           // stride 132 -> conflict-free
  const int m0  = blockIdx.x * 16;
  const int tid = threadIdx.x;

  // stage A panel 16x128 -> LDS (2048 floats, 8 per thread)
  {
    int r = tid >> 4;            // 0..15
    int c = (tid & 15) * 8;      // 0..120
    const float4* src = (const float4*)(A + (size_t)(m0 + r) * IN_CH + c);
    float4 a0 = src[0], a1 = src[1];
    float* d = lds + r * 132 + c;
    d[0] = a0.x; d[1] = a0.y; d[2] = a0.z; d[3] = a0.w;
    d[4] = a1.x; d[5] = a1.y; d[6] = a1.z; d[7] = a1.w;
  }
  __syncthreads();

  const int wave = tid >> 5;
  const int lane = tid & 31;
  const int half = lane >> 4;    // lane half selects K 0,1 vs 2,3
  const int lr   = lane & 15;
  const int n0   = wave * 32;    // this wave's two tiles: n0, n0+16

  v8f c0 = {}; v8f c1 = {};
  const float* lrow = lds + lr * 132 + 2 * half;

  #pragma unroll 4
  for (int k0 = 0; k0 < IN_CH; k0 += 4) {
    v2f a;                                   // A frag: {K,K+1} per lane-half
    a.x = lrow[k0];
    a.y = lrow[k0 + 1];
    const float* brow = B + (size_t)(k0 + 2 * half) * TOT_CH + lr;
    v2f b0, b1;                              // B frag: VGPR0=K0/K2, VGPR1=K1/K3
    b0.x = brow[n0];          b0.y = brow[TOT_CH + n0];
    b1.x = brow[n0 + 16];     b1.y = brow[TOT_CH + n0 + 16];
    c0 = __builtin_amdgcn_wmma_f32_16x16x4_f32(false, a, false, b0, (short)0, c0, false, false);
    c1 = __builtin_amdgcn_wmma_f32_16x16x4_f32(false, a, false, b1, (short)0, c1, false, false);
  }

  #pragma unroll
  for (int r = 0; r < 8; ++r) {              // C/D layout: VGPR r -> row r (+8 for hi half)
    int m = m0 + r + 8 * half;
    X[(size_t)m * TOT_CH + n0 + lr]      = c0[r];
    X[(size_t)m * TOT_CH + n0 + 16 + lr] = c1[r];
  }
}

// ---------------------------------------------------------------------------
// Kernel 2: per-node attention logits. One wave per node; lanes 0-15 head 0,
// lanes 16-31 head 1; shfl-xor reduction within each 16-lane half.
// ---------------------------------------------------------------------------
__global__ __launch_bounds__(256) void gat_logits(const float* __restrict__ X,
                                                  const float* __restrict__ att_s,
                                                  const float* __restrict__ att_d,
                                                  float* __restrict__ a_src,
                                                  float* __restrict__ a_dst) {
  int wave = threadIdx.x >> 5;
  int lane = threadIdx.x & 31;
  int node = blockIdx.x * 8 + wave;
  if (node >= N_NODES) return;
  int h  = lane >> 4;
  int lh = lane & 15;
  int off = h * OUT_CH + lh * 8;
  const float4* xs = (const float4*)(X + (size_t)node * TOT_CH + off);
  const float4* as = (const float4*)(att_s + off);
  const float4* ad = (const float4*)(att_d + off);
  float4 x0 = xs[0], x1 = xs[1];
  float4 s0 = as[0], s1 = as[1];
  float4 d0 = ad[0], d1 = ad[1];
  float ss = x0.x*s0.x + x0.y*s0.y + x0.z*s0.z + x0.w*s0.w
           + x1.x*s1.x + x1.y*s1.y + x1.z*s1.z + x1.w*s1.w;
  float ds = x0.x*d0.x + x0.y*d0.y + x0.z*d0.z + x0.w*d0.w
           + x1.x*d1.x + x1.y*d1.y + x1.z*d1.z + x1.w*d1.w;
  #pragma unroll
  for (int m = 1; m <= 8; m <<= 1) {   // stays inside each 16-lane half
    ss += __shfl_xor(ss, m, 32);
    ds += __shfl_xor(ds, m, 32);
  }
  if (lh == 0) {
    a_src[node * HEADS + h] = ss;
    a_dst[node * HEADS + h] = ds;
  }
}

// ---------------------------------------------------------------------------
// Kernel 3: init accumulators
// ---------------------------------------------------------------------------
__global__ void gat_init(float* __restrict__ out,
                         float* __restrict__ seg_max,
                         float* __restrict__ seg_sum) {
  int i = blockIdx.x * blockDim.x + threadIdx.x;
  if (i < N_NODES * TOT_CH) out[i] = 0.0f;
  if (i < N_NODES * HEADS) { seg_max[i] = -__builtin_inff(); seg_sum[i] = 0.0f; }
}

// ---------------------------------------------------------------------------
// Kernel 4: segment max of leaky-relu logits over destinations
// ---------------------------------------------------------------------------
__global__ void gat_edge_max(const int* __restrict__ ei,
                             const float* __restrict__ a_src,
                             const float* __restrict__ a_dst,
                             float* __restrict__ seg_max) {
  int i = blockIdx.x * blockDim.x + threadIdx.x;
  if (i >= N_EDGES + N_NODES) return;
  int s, d;
  if (i < N_EDGES) { s = ei[i]; d = ei[N_EDGES + i]; } else { s = d = i - N_EDGES; }
  #pragma unroll
  for (int h = 0; h < HEADS; ++h) {
    float v = lrelu(a_src[s * HEADS + h] + a_dst[d * HEADS + h]);
    atomicMaxF(&seg_max[d * HEADS + h], v);
  }
}

// ---------------------------------------------------------------------------
// Kernel 5: segment sum of exp(e - max)
// ---------------------------------------------------------------------------
__global__ void gat_edge_sum(const int* __restrict__ ei,
                             const float* __restrict__ a_src,
                             const float* __restrict__ a_dst,
                             const float* __restrict__ seg_max,
                             float* __restrict__ seg_sum) {
  int i = blockIdx.x * blockDim.x + threadIdx.x;
  if (i >= N_EDGES + N_NODES) return;
  int s, d;
  if (i < N_EDGES) { s = ei[i]; d = ei[N_EDGES + i]; } else { s = d = i - N_EDGES; }
  #pragma unroll
  for (int h = 0; h < HEADS; ++h) {
    float v = lrelu(a_src[s * HEADS + h] + a_dst[d * HEADS + h]);
    atomicAdd(&seg_sum[d * HEADS + h], expf(v - seg_max[d * HEADS + h]));
  }
}

// ---------------------------------------------------------------------------
// Kernel 6: weighted scatter-sum. One wave per edge; lane covers 8 channels
// (lane>>4 selects the head). x[src] rows are L2-resident (51 MB < 192 MB L2).
// ---------------------------------------------------------------------------
__global__ __launch_bounds__(256) void gat_aggregate(const int* __restrict__ ei,
                                                     const float* __restrict__ X,
                                                     const float* __restrict__ a_src,
                                                     const float* __restrict__ a_dst,
                                                     const float* __restrict__ seg_max,
                                                     const float* __restrict__ seg_sum,
                                                     float* __restrict__ out) {
  int wave = threadIdx.x >> 5;
  int lane = threadIdx.x & 31;
  int e = blockIdx.x * 8 + wave;
  if (e >= N_EDGES + N_NODES) return;
  int s, d;
  if (e < N_EDGES) { s = ei[e]; d = ei[N_EDGES + e]; } else { s = d = e - N_EDGES; }
  int h = lane >> 4;
  float lv    = lrelu(a_src[s * HEADS + h] + a_dst[d * HEADS + h]);
  float alpha = expf(lv - seg_max[d * HEADS + h]) / (seg_sum[d * HEADS + h] + 1e-16f);
  const float4* xs = (const float4*)(X + (size_t)s * TOT_CH + lane * 8);
  float4 v0 = xs[0], v1 = xs[1];
  float* o = out + (size_t)d * TOT_CH + lane * 8;
  atomicAdd(o + 0, alpha * v0.x); atomicAdd(o + 1, alpha * v0.y);
  atomicAdd(o + 2, alpha * v0.z); atomicAdd(o + 3, alpha * v0.w);
  atomicAdd(o + 4, alpha * v1.x); atomicAdd(o + 5, alpha * v1.y);
  atomicAdd(o + 6, alpha * v1.z); atomicAdd(o + 7, alpha * v1.w);
}

// ---------------------------------------------------------------------------
// Kernel 7: out = elu(out + bias), in place
// ---------------------------------------------------------------------------
__global__ void gat_finalize(float* __restrict__ out, const float* __restrict__ bias) {
  int i = blockIdx.x * blockDim.x + threadIdx.x;
  if (i >= N_NODES * TOT_CH) return;
  float v = out[i] + bias[i & (TOT_CH - 1)];
  out[i] = v > 0.0f ? v : expm1f(v);
}

// ---------------------------------------------------------------------------
extern "C" void kernel_launch(void* const* d_in, const int* in_sizes, int n_in,
                              void* d_out, int out_size, void* d_ws, size_t ws_size,
                              hipStream_t stream) {
  const float* h_node = (const float*)d_in[0];
  const int*   ei     = (const int*)  d_in[1];   // [2, E] int32
  const float* W      = (const float*)d_in[2];
  const float* att_s  = (const float*)d_in[3];
  const float* att_d  = (const float*)d_in[4];
  const float* bias   = (const float*)d_in[5];
  float* out = (float*)d_out;

  // workspace layout (floats): X | a_src | a_dst | seg_max | seg_sum  (~52.8 MB)
  float* ws      = (float*)d_ws;
  float* X       = ws;                                   // N*256
  float* a_src   = X + (size_t)N_NODES * TOT_CH;         // N*2
  float* a_dst   = a_src + (size_t)N_NODES * HEADS;      // N*2
  float* seg_max = a_dst + (size_t)N_NODES * HEADS;      // N*2
  float* seg_sum = seg_max + (size_t)N_NODES * HEADS;    // N*2

  const int EN = N_EDGES + N_NODES;

  gat_gemm     <<<N_NODES / 16, 256, 0, stream>>>(h_node, W, X);
  gat_logits   <<<(N_NODES + 7) / 8, 256, 0, stream>>>(X, att_s, att_d, a_src, a_dst);
  gat_init     <<<(N_NODES * TOT_CH + 255) / 256, 256, 0, stream>>>(out, seg_max, seg_sum);
  gat_edge_max <<<(EN + 255) / 256, 256, 0, stream>>>(ei, a_src, a_dst, seg_max);
  gat_edge_sum <<<(EN + 255) / 256, 256, 0, stream>>>(ei, a_src, a_dst, seg_max, seg_sum);
  gat_aggregate<<<(EN + 7) / 8, 256, 0, stream>>>(ei, X, a_src, a_dst, seg_max, seg_sum, out);
  gat_finalize <<<(N_NODES * TOT_CH + 255) / 256, 256, 0, stream>>>(out, bias);
}